// BboxSegEnsembler1Case_3427383902217
// MI455X (gfx1250) — compile-verified
//
#include <hip/hip_runtime.h>
#include <stdint.h>

#define N_PROP   200000
#define K_PRE    1000
#define CAND_MAX 2048
#define NBINS    32768

// ---- workspace layout (bytes) ----
#define WS_HIST 0           // NBINS u32            = 131072
#define WS_CNT  131072      // 64 u32: [0]=candCount [1]=cutBin [8..39]=validMask
#define WS_CAND 131328      // CAND_MAX u64         = 16384
#define WS_PBOX 147712      // 1024*8 f32 (offset boxes, score, label) = 32768
#define WS_POUT 180480      // 1024*6 f32 (clipped, no offset)         = 24576
#define WS_MASK 205056      // K_PRE*32 u32                            = 128000
// total 333056 bytes

#define SEG_ELEMS 14155776  // 2*192*192*192
#define DETS_OFF  14155776
#define LAB_OFF   14156576

// ---- CDNA5 async global->LDS (ASYNCcnt path) ----
// clang prototype (from round-1 diagnostic): arg0 = int4 addrspace(1)*,
// arg1 = int4 addrspace(3)*, then two immediate ints (offset, cpol).
typedef int v4i_t __attribute__((vector_size(16)));
typedef __attribute__((address_space(1))) v4i_t gv4i_t;
typedef __attribute__((address_space(3))) v4i_t lv4i_t;

__device__ __forceinline__ void async_g2l_b128(const void* g, void* l) {
#if __has_builtin(__builtin_amdgcn_global_load_async_to_lds_b128)
  __builtin_amdgcn_global_load_async_to_lds_b128(
      (gv4i_t*)(v4i_t*)(void*)(uintptr_t)g,   // generic -> AS1 (no-op addrspacecast)
      (lv4i_t*)(v4i_t*)l,                     // generic -> AS3 (LDS)
      0, 0);
#else
  unsigned lofs = (unsigned)(uintptr_t)l;   // generic LDS addr low 32 bits == LDS offset
  unsigned long long ga = (unsigned long long)(uintptr_t)g;
  asm volatile("global_load_async_to_lds_b128 %0, %1, off"
               :: "v"(lofs), "v"(ga) : "memory");
#endif
}
__device__ __forceinline__ void wait_async0() {
#if __has_builtin(__builtin_amdgcn_s_wait_asynccnt)
  __builtin_amdgcn_s_wait_asynccnt(0);
#else
  asm volatile("s_wait_asynccnt 0x0" ::: "memory");
#endif
}

// =====================================================================
// Kernel 1: seg ensemble.  out[c,x,y,z] = tile_last * w_last / sum(w_cov)
// last covering tile per dim = largest origin in {0,48,96} covering coord.
// float4 over z (segments of 48 are 4-aligned).
// =====================================================================
__global__ void k_seg(const float* __restrict__ tiles,
                      const float* __restrict__ wgt,
                      float* __restrict__ out) {
  long tid = (long)blockIdx.x * blockDim.x + threadIdx.x;
  const long total = 2L * 192 * 192 * 48;
  if (tid >= total) return;
  int z4 = (int)(tid % 48); long r = tid / 48;
  int y  = (int)(r % 192);  r /= 192;
  int x  = (int)(r % 192);  int c = (int)(r / 192);
  int z  = z4 * 4;

  int lx = x >= 96 ? 96 : (x >= 48 ? 48 : 0);
  int ly = y >= 96 ? 96 : (y >= 48 ? 48 : 0);
  int lz = z >= 96 ? 96 : (z >= 48 ? 48 : 0);
  int nx = (x >= 48 && x < 144) ? 2 : 1;
  int ny = (y >= 48 && y < 144) ? 2 : 1;
  int nz = (z >= 48 && z < 144) ? 2 : 1;
  int t  = (lx / 48) * 9 + (ly / 48) * 3 + (lz / 48);

  long toff = ((((long)t * 2 + c) * 96 + (x - lx)) * 96 + (y - ly)) * 96 + (z - lz);
  float4 v  = *(const float4*)(tiles + toff);
  float4 wl = *(const float4*)(wgt + (((long)(x - lx) * 96 + (y - ly)) * 96 + (z - lz)));
  float4 num; num.x = v.x * wl.x; num.y = v.y * wl.y; num.z = v.z * wl.z; num.w = v.w * wl.w;

  float4 den = make_float4(0.f, 0.f, 0.f, 0.f);
  int ox0 = (nx == 2) ? lx - 48 : lx;
  int oy0 = (ny == 2) ? ly - 48 : ly;
  int oz0 = (nz == 2) ? lz - 48 : lz;
  for (int a = 0; a < nx; ++a)
    for (int b = 0; b < ny; ++b)
      for (int e = 0; e < nz; ++e) {
        int dx = x - (ox0 + 48 * a), dy = y - (oy0 + 48 * b), dz = z - (oz0 + 48 * e);
        float4 w4 = *(const float4*)(wgt + (((long)dx * 96 + dy) * 96 + dz));
        den.x += w4.x; den.y += w4.y; den.z += w4.z; den.w += w4.w;
      }

  float4 res;
  res.x = num.x / den.x; res.y = num.y / den.y;
  res.z = num.z / den.z; res.w = num.w / den.w;
  long oo = (((long)c * 192 + x) * 192 + y) * 192L + z;
  *(float4*)(out + oo) = res;
}

// =====================================================================
// Top-1000 selection: histogram on monotone float bits (scores >= 0)
// =====================================================================
__global__ void k_hist(const float* __restrict__ bbox, unsigned* __restrict__ hist) {
  int n = blockIdx.x * blockDim.x + threadIdx.x;
  if (n >= N_PROP) return;
  unsigned bits = __float_as_uint(bbox[(long)n * 7 + 6]);
  unsigned bin = bits >> 15; if (bin >= NBINS) bin = NBINS - 1;
  atomicAdd(&hist[bin], 1u);
}

__global__ void __launch_bounds__(1024) k_cut(const unsigned* __restrict__ hist,
                                              unsigned* __restrict__ cnts) {
  __shared__ unsigned chunk[1024];
  unsigned t = threadIdx.x;
  unsigned s = 0;
  for (int b = 0; b < 32; ++b) s += hist[t * 32 + b];
  chunk[t] = s;
  __syncthreads();
  if (t == 0) {
    unsigned acc = 0; unsigned cut = 0;
    for (int c = 1023; c >= 0; --c) {
      if (acc + chunk[c] >= K_PRE) {
        for (int b = 31; b >= 0; --b) {
          acc += hist[c * 32 + b];
          if (acc >= K_PRE) { cut = (unsigned)(c * 32 + b); break; }
        }
        break;
      }
      acc += chunk[c];
    }
    cnts[1] = cut;
  }
}

__global__ void k_compact(const float* __restrict__ bbox,
                          unsigned* __restrict__ cnts,
                          unsigned long long* __restrict__ cand) {
  int n = blockIdx.x * blockDim.x + threadIdx.x;
  if (n >= N_PROP) return;
  unsigned bits = __float_as_uint(bbox[(long)n * 7 + 6]);
  unsigned bin = bits >> 15; if (bin >= NBINS) bin = NBINS - 1;
  if (bin >= cnts[1]) {
    unsigned pos = atomicAdd(&cnts[0], 1u);
    if (pos < CAND_MAX)
      cand[pos] = ((unsigned long long)bits << 32) | (unsigned)(~(unsigned)n);
  }
}

// =====================================================================
// Single block: async-stage candidates to LDS, bitonic-sort 2048 keys
// descending (== jax top_k order incl. tie-break), then build box table.
// =====================================================================
__global__ void __launch_bounds__(1024) k_sortprep(
    const float* __restrict__ bbox, const int* __restrict__ labels,
    const unsigned long long* __restrict__ cand, unsigned* __restrict__ cnts,
    float* __restrict__ pbox, float* __restrict__ pout) {
  __shared__ unsigned long long keys[CAND_MAX];
  unsigned t = threadIdx.x;

  // async 16KB stage: each thread one b128 (2 keys)
  async_g2l_b128(&cand[t * 2], &keys[t * 2]);
  wait_async0();
  __syncthreads();

  unsigned cc = cnts[0]; if (cc > CAND_MAX) cc = CAND_MAX;
  if (t >= cc)            keys[t] = 0ull;
  if (t + 1024 >= cc)     keys[t + 1024] = 0ull;

  // bitonic sort, descending
  for (unsigned k = 2; k <= CAND_MAX; k <<= 1)
    for (unsigned j = k >> 1; j > 0; j >>= 1) {
      __syncthreads();
      for (unsigned i = t; i < CAND_MAX; i += 1024) {
        unsigned p = i ^ j;
        if (p > i) {
          bool up = ((i & k) == 0);
          unsigned long long a = keys[i], b = keys[p];
          bool sw = up ? (a < b) : (a > b);
          if (sw) { keys[i] = b; keys[p] = a; }
        }
      }
    }
  __syncthreads();

  // prep: clip, validity, class offset
  unsigned long long key = keys[t];
  bool real = (t < K_PRE) && (key != 0ull);
  float c[6] = {0.f,0.f,0.f,0.f,0.f,0.f};
  float sc = 0.f; int lab = 0; bool valid = false;
  if (real) {
    unsigned n = ~(unsigned)(key & 0xFFFFFFFFull);
    sc  = bbox[(long)n * 7 + 6];
    lab = labels[n];
    valid = sc > 0.01f;
    for (int d = 0; d < 6; ++d) {
      float v = bbox[(long)n * 7 + d];
      c[d] = fminf(fmaxf(v, 0.f), 192.f);
    }
    for (int d = 0; d < 3; ++d) valid = valid && (c[d + 3] - c[d] >= 0.01f);
  }
  float off = (float)lab * 384.0f;   // CLASS_OFFSET
  for (int d = 0; d < 6; ++d) {
    pbox[t * 8 + d] = real ? (c[d] + off) : 0.f;
    pout[t * 6 + d] = real ? c[d] : 0.f;
  }
  pbox[t * 8 + 6] = sc;
  pbox[t * 8 + 7] = (float)lab;
  if (valid) atomicOr(&cnts[8 + (t >> 5)], 1u << (t & 31));
}

// =====================================================================
// IoU suppression mask: block = row i, lanes = columns, boxes in LDS
// (async-staged), ballot_w32 packs predicate words.
// =====================================================================
__global__ void __launch_bounds__(1024) k_mask(const float* __restrict__ pbox,
                                               unsigned* __restrict__ masks) {
  __shared__ float sb[1024 * 8];
  unsigned t = threadIdx.x;
  async_g2l_b128(pbox + t * 8,     &sb[t * 8]);
  async_g2l_b128(pbox + t * 8 + 4, &sb[t * 8 + 4]);
  wait_async0();
  __syncthreads();

  int i = blockIdx.x;
  const float* A = &sb[(unsigned)i * 8];
  const float* B = &sb[t * 8];
  float lo0 = fmaxf(A[0], B[0]), hi0 = fminf(A[3], B[3]);
  float lo1 = fmaxf(A[1], B[1]), hi1 = fminf(A[4], B[4]);
  float lo2 = fmaxf(A[2], B[2]), hi2 = fminf(A[5], B[5]);
  float inter = fmaxf(hi0 - lo0, 0.f) * fmaxf(hi1 - lo1, 0.f) * fmaxf(hi2 - lo2, 0.f);
  float va = (A[3] - A[0]) * (A[4] - A[1]) * (A[5] - A[2]);
  float vb = (B[3] - B[0]) * (B[4] - B[1]) * (B[5] - B[2]);
  float iou = inter / fmaxf(va + vb - inter, 1e-6f);
  bool pred = ((int)t > i) && (iou > 0.1f);

#if __has_builtin(__builtin_amdgcn_ballot_w32)
  unsigned word = __builtin_amdgcn_ballot_w32(pred);
  if ((t & 31u) == 0u) masks[(unsigned)i * 32 + (t >> 5)] = word;
#else
  __shared__ unsigned shw[32];
  if (t < 32) shw[t] = 0u;
  __syncthreads();
  if (pred) atomicOr(&shw[t >> 5], 1u << (t & 31));
  __syncthreads();
  if (t < 32) masks[(unsigned)i * 32 + t] = shw[t];
#endif
}

// =====================================================================
// Serial greedy NMS on ONE wave32: lane L owns removal word L (1024 bits
// total), rows prefetched from L2, cross-lane alive check via LDS mirror.
// Then emit top-100 dets (+labels as float).
// =====================================================================
__global__ void __launch_bounds__(32) k_nms(const float* __restrict__ pbox,
                                            const float* __restrict__ pout,
                                            const unsigned* __restrict__ cnts,
                                            const unsigned* __restrict__ masks,
                                            float* __restrict__ out_dets,
                                            float* __restrict__ out_labels) {
  int lane = threadIdx.x;
  volatile __shared__ unsigned sh_remv[32];
  __shared__ int sh_fidx[100];
  __shared__ int sh_cnt;

  unsigned validw = cnts[8 + lane];
  unsigned remv = ~validw;
  sh_remv[lane] = remv;
  __syncthreads();

  unsigned row = masks[lane];                       // row 0
  for (int i = 0; i < K_PRE; ++i) {
    unsigned nrow = (i + 1 < K_PRE) ? masks[(unsigned)(i + 1) * 32 + lane] : 0u;
    unsigned rw = sh_remv[i >> 5];
    if (((rw >> (i & 31)) & 1u) == 0u) {            // row i still alive -> suppress
      remv |= row;
      sh_remv[lane] = remv;
    }
    row = nrow;
  }
  __syncthreads();

  if (lane == 0) {
    int cnt = 0;
    for (int j = 0; j < K_PRE && cnt < 100; ++j) {
      if (((sh_remv[j >> 5] >> (j & 31)) & 1u) == 0u) sh_fidx[cnt++] = j;
    }
    sh_cnt = cnt;
  }
  __syncthreads();

  int cnt = sh_cnt;
  for (int r = lane; r < 100; r += 32) {
    if (r < cnt) {
      int j = sh_fidx[r];
      float sc = pbox[j * 8 + 6];
      int lab = (int)pbox[j * 8 + 7];
      for (int d = 0; d < 6; ++d) out_dets[r * 8 + d] = pout[j * 6 + d];
      out_dets[r * 8 + 6] = sc;
      out_dets[r * 8 + 7] = sc;
      out_labels[r] = (float)lab;
    } else {
      for (int d = 0; d < 8; ++d) out_dets[r * 8 + d] = 0.f;
      out_labels[r] = -1.f;
    }
  }
}

// =====================================================================
extern "C" void kernel_launch(void* const* d_in, const int* in_sizes, int n_in,
                              void* d_out, int out_size, void* d_ws, size_t ws_size,
                              hipStream_t stream) {
  const float* tiles  = (const float*)d_in[0];
  const float* wgt    = (const float*)d_in[1];
  const float* bbox   = (const float*)d_in[2];
  const int*   labels = (const int*)d_in[3];
  // d_in[4] tile_origins: analytic {0,48,96}^3 grid, not needed at runtime.

  float* out = (float*)d_out;
  unsigned char* ws = (unsigned char*)d_ws;
  unsigned*           hist = (unsigned*)(ws + WS_HIST);
  unsigned*           cnts = (unsigned*)(ws + WS_CNT);
  unsigned long long* cand = (unsigned long long*)(ws + WS_CAND);
  float*              pbox = (float*)(ws + WS_PBOX);
  float*              pout = (float*)(ws + WS_POUT);
  unsigned*           mask = (unsigned*)(ws + WS_MASK);

  // reset histogram + counters + valid mask every launch (graph-capturable)
  (void)hipMemsetAsync(ws, 0, WS_CAND, stream);

  k_seg<<<dim3(13824), dim3(256), 0, stream>>>(tiles, wgt, out);

  k_hist<<<dim3((N_PROP + 255) / 256), dim3(256), 0, stream>>>(bbox, hist);
  k_cut<<<dim3(1), dim3(1024), 0, stream>>>(hist, cnts);
  k_compact<<<dim3((N_PROP + 255) / 256), dim3(256), 0, stream>>>(bbox, cnts, cand);
  k_sortprep<<<dim3(1), dim3(1024), 0, stream>>>(bbox, labels, cand, cnts, pbox, pout);
  k_mask<<<dim3(K_PRE), dim3(1024), 0, stream>>>(pbox, mask);
  k_nms<<<dim3(1), dim3(32), 0, stream>>>(pbox, pout, cnts, mask,
                                          out + DETS_OFF, out + LAB_OFF);
}